// GATNet_1726576853645
// MI455X (gfx1250) — compile-verified
//
#include <hip/hip_runtime.h>
#include <cstdint>
#include <climits>

#define FEAT 64
#define NHEAD 8
#define CH 16
#define HC 128
#define NEG_SLOPE 0.2f

typedef float v2f __attribute__((ext_vector_type(2)));
typedef float v8f __attribute__((ext_vector_type(8)));

// Monotonic float<->int encoding so integer atomicMax == float max (handles negatives).
__device__ __forceinline__ int enc_f32(float f) {
  int i = __float_as_int(f);
  return (i >= 0) ? i : (i ^ 0x7FFFFFFF);
}
__device__ __forceinline__ float dec_f32(int i) {
  return __int_as_float((i >= 0) ? i : (i ^ 0x7FFFFFFF));
}

// ---------------------------------------------------------------------------
// GEMM: Hout[N,128] = X[N,K] @ W[K,128] using V_WMMA_F32_16X16X4_F32.
// Block = 256 threads = 8 wave32 waves. Block handles 16 rows x 128 cols;
// each wave owns one 16-column slab (== one attention head, since C==16).
// X tile staged in LDS (padded: row stride K+4 => bank stride 4 mod 64,
// 16 lanes hit 16 distinct banks).
// A layout: lane l, vgpr v -> M=l%16, K = k0 + 2*(l/16) + v
// B layout: lane l, vgpr v -> N=l%16, K = k0 + 2*(l/16) + v
// C/D     : lane l, vgpr v -> N=l%16, M = v + 8*(l/16)
// ---------------------------------------------------------------------------
template <int K>
__launch_bounds__(256)
__global__ void gemm_wmma_k(const float* __restrict__ X,
                            const float* __restrict__ W,
                            float* __restrict__ Hout) {
  constexpr int KP = K + 4;
  __shared__ float xs[16 * KP];
  const int tid = threadIdx.x;
  const int row0 = blockIdx.x * 16;

  for (int i = tid; i < 16 * K; i += 256) {
    const int r = i / K;
    const int c = i - r * K;
    xs[r * KP + c] = X[(size_t)(row0 + r) * K + c];
  }
  __syncthreads();

  const int lane = tid & 31;
  const int wave = tid >> 5;
  const int col0 = wave * 16;          // 8 waves * 16 = 128 output cols
  const int m    = lane & 15;
  const int kh   = (lane >> 4) << 1;   // 0 or 2

  v8f acc = {};
  const float* xrow = &xs[m * KP + kh];
  const float* wcol = W + col0 + m;

#pragma unroll
  for (int k0 = 0; k0 < K; k0 += 4) {
    v2f a, b;
    a.x = xrow[k0];
    a.y = xrow[k0 + 1];
    b.x = wcol[(size_t)(k0 + kh) * HC];
    b.y = wcol[(size_t)(k0 + kh + 1) * HC];
    acc = __builtin_amdgcn_wmma_f32_16x16x4_f32(false, a, false, b,
                                                (short)0, acc, false, false);
  }

  const int rbase = row0 + ((lane >> 4) << 3);
  float* op = Hout + (size_t)rbase * HC + col0 + m;
#pragma unroll
  for (int v = 0; v < 8; ++v) op[(size_t)v * HC] = acc[v];
}

// ---------------------------------------------------------------------------
// Per-(node,head) attention logits: als = <h[n,h,:], a_src[h,:]>, same for dst.
// ---------------------------------------------------------------------------
__global__ void attn_coeff(const float* __restrict__ Hf,
                           const float* __restrict__ asrc,
                           const float* __restrict__ adst,
                           float* __restrict__ als, float* __restrict__ ald,
                           int total /* N*NHEAD */) {
  const int idx = blockIdx.x * blockDim.x + threadIdx.x;
  if (idx >= total) return;
  const int hh = idx & (NHEAD - 1);
  const float4* hv  = (const float4*)(Hf + (size_t)idx * CH);
  const float4* as4 = (const float4*)(asrc + hh * CH);
  const float4* ad4 = (const float4*)(adst + hh * CH);
  float s = 0.f, d = 0.f;
#pragma unroll
  for (int q = 0; q < 4; ++q) {
    const float4 h4 = hv[q], a4 = as4[q], b4 = ad4[q];
    s += h4.x * a4.x + h4.y * a4.y + h4.z * a4.z + h4.w * a4.w;
    d += h4.x * b4.x + h4.y * b4.y + h4.z * b4.z + h4.w * b4.w;
  }
  als[idx] = s;
  ald[idx] = d;
}

__global__ void fill_int(int* __restrict__ p, int val, int n) {
  const int i = blockIdx.x * blockDim.x + threadIdx.x;
  if (i < n) p[i] = val;
}

__device__ __forceinline__ void edge_sd(const int* __restrict__ ei, int e, int E,
                                        int& s, int& d) {
  if (e < E) { s = ei[e]; d = ei[E + e]; }      // real edge
  else       { s = e - E; d = s; }              // self loop
}

__device__ __forceinline__ float lrelu(float v) {
  return (v > 0.f) ? v : NEG_SLOPE * v;
}

// Pass 1: segment max of leaky_relu(als[src]+ald[dst]) per (dst, head).
__global__ void edge_max_k(const int* __restrict__ ei, const float* __restrict__ als,
                           const float* __restrict__ ald, int* __restrict__ menc,
                           int E, int total /* nedges*NHEAD */) {
  const int t = blockIdx.x * blockDim.x + threadIdx.x;
  if (t >= total) return;
  const int e = t >> 3, hh = t & 7;
  int s, d;
  edge_sd(ei, e, E, s, d);
  const float v = lrelu(als[s * NHEAD + hh] + ald[d * NHEAD + hh]);
  atomicMax(menc + d * NHEAD + hh, enc_f32(v));
}

// Pass 2: denom[dst,h] += exp(e - m[dst,h])
__global__ void edge_denom_k(const int* __restrict__ ei, const float* __restrict__ als,
                             const float* __restrict__ ald, const int* __restrict__ menc,
                             float* __restrict__ denom, int E, int total) {
  const int t = blockIdx.x * blockDim.x + threadIdx.x;
  if (t >= total) return;
  const int e = t >> 3, hh = t & 7;
  int s, d;
  edge_sd(ei, e, E, s, d);
  const int o = d * NHEAD + hh;
  const float v = lrelu(als[s * NHEAD + hh] + ald[o]);
  atomicAdd(denom + o, expf(v - dec_f32(menc[o])));
}

// Pass 3: aggr[dst,:] += h[src,:] * softmax_weight.  One wave per edge,
// 4 floats per lane (lane -> head = lane/4, chans lane*4 .. lane*4+3).
__global__ void edge_aggr_k(const int* __restrict__ ei, const float* __restrict__ als,
                            const float* __restrict__ ald, const int* __restrict__ menc,
                            const float* __restrict__ denom, const float* __restrict__ Hf,
                            float* __restrict__ aggr, int E, int nedges) {
  const int t = blockIdx.x * blockDim.x + threadIdx.x;
  const int e = t >> 5;
  if (e >= nedges) return;
  const int lane = t & 31;
  int s, d;
  edge_sd(ei, e, E, s, d);
  const int hh = lane >> 2;
  const int o = d * NHEAD + hh;
  const float v = lrelu(als[s * NHEAD + hh] + ald[o]);
  const float w = expf(v - dec_f32(menc[o])) / denom[o];
  const float4 hv = *(const float4*)(Hf + (size_t)s * HC + lane * 4);
  float* op = aggr + (size_t)d * HC + lane * 4;
  atomicAdd(op + 0, hv.x * w);
  atomicAdd(op + 1, hv.y * w);
  atomicAdd(op + 2, hv.z * w);
  atomicAdd(op + 3, hv.w * w);
}

__global__ void bias_elu_k(const float* __restrict__ aggr, const float* __restrict__ bias,
                           float* __restrict__ act, int total) {
  const int i = blockIdx.x * blockDim.x + threadIdx.x;
  if (i >= total) return;
  const float v = aggr[i] + bias[i & (HC - 1)];
  act[i] = (v > 0.f) ? v : expm1f(v);
}

__global__ void pool_k(const float* __restrict__ act, const int* __restrict__ batch,
                       float* __restrict__ g, int total) {
  const int i = blockIdx.x * blockDim.x + threadIdx.x;
  if (i >= total) return;
  const int n = i >> 7, k = i & (HC - 1);
  atomicAdd(g + (size_t)batch[n] * HC + k, act[i]);
}

// Fused MLP head: out[row] = elu(g[row]@W1 + b1) @ W2 + b2.  One wave per row;
// lanes 0..15 each own one hidden unit; reduce with shuffles (width 16).
__global__ void mlp_k(const float* __restrict__ g, const float* __restrict__ w1,
                      const float* __restrict__ b1, const float* __restrict__ w2,
                      const float* __restrict__ b2, float* __restrict__ out) {
  const int row = blockIdx.x;
  const int lane = threadIdx.x;
  const int c = lane & 15;
  const float* gr = g + (size_t)row * HC;
  float acc = 0.f;
  for (int k = 0; k < HC; ++k) acc += gr[k] * w1[k * CH + c];
  acc += b1[c];
  acc = (acc > 0.f) ? acc : expm1f(acc);
  float t = acc * w2[c];
#pragma unroll
  for (int mask = 8; mask >= 1; mask >>= 1) t += __shfl_xor(t, mask, 16);
  if (lane == 0) out[row] = t + b2[0];
}

// ---------------------------------------------------------------------------
extern "C" void kernel_launch(void* const* d_in, const int* in_sizes, int n_in,
                              void* d_out, int out_size, void* d_ws, size_t ws_size,
                              hipStream_t stream) {
  const float* x     = (const float*)d_in[0];
  const int*   ei    = (const int*)d_in[1];
  const int*   batch = (const int*)d_in[2];
  const float* W1    = (const float*)d_in[3];
  const float* a1s   = (const float*)d_in[4];
  const float* a1d   = (const float*)d_in[5];
  const float* b1    = (const float*)d_in[6];
  const float* W2    = (const float*)d_in[7];
  const float* a2s   = (const float*)d_in[8];
  const float* a2d   = (const float*)d_in[9];
  const float* b2    = (const float*)d_in[10];
  const float* l1w   = (const float*)d_in[11];
  const float* l1b   = (const float*)d_in[12];
  const float* l2w   = (const float*)d_in[13];
  const float* l2b   = (const float*)d_in[14];
  float* out = (float*)d_out;

  const int Nn = in_sizes[0] / FEAT;   // 50000 (divisible by 16)
  const int E  = in_sizes[1] / 2;      // 1600000
  const int Bb = out_size;             // 512
  const int nedges = E + Nn;

  char* ws = (char*)d_ws;
  float* hbuf  = (float*)ws;  ws += (size_t)Nn * HC * 4;
  float* aggr  = (float*)ws;  ws += (size_t)Nn * HC * 4;
  float* act   = (float*)ws;  ws += (size_t)Nn * HC * 4;
  float* als   = (float*)ws;  ws += (size_t)Nn * NHEAD * 4;
  float* ald   = (float*)ws;  ws += (size_t)Nn * NHEAD * 4;
  int*   menc  = (int*)ws;    ws += (size_t)Nn * NHEAD * 4;
  float* denom = (float*)ws;  ws += (size_t)Nn * NHEAD * 4;
  float* g     = (float*)ws;  ws += (size_t)Bb * HC * 4;

  const int nhTot = Nn * NHEAD;
  const int ehTot = nedges * NHEAD;

  for (int layer = 0; layer < 3; ++layer) {
    const float* Xin = (layer == 0) ? x : act;
    const float* Wl  = (layer == 0) ? W1  : (W2  + (size_t)(layer - 1) * HC * HC);
    const float* asl = (layer == 0) ? a1s : (a2s + (size_t)(layer - 1) * NHEAD * CH);
    const float* adl = (layer == 0) ? a1d : (a2d + (size_t)(layer - 1) * NHEAD * CH);
    const float* bl  = (layer == 0) ? b1  : (b2  + (size_t)(layer - 1) * HC);

    if (layer == 0)
      gemm_wmma_k<FEAT><<<Nn / 16, 256, 0, stream>>>(Xin, Wl, hbuf);
    else
      gemm_wmma_k<HC><<<Nn / 16, 256, 0, stream>>>(Xin, Wl, hbuf);

    attn_coeff<<<(nhTot + 255) / 256, 256, 0, stream>>>(hbuf, asl, adl, als, ald, nhTot);

    fill_int<<<(nhTot + 255) / 256, 256, 0, stream>>>(menc, INT_MIN, nhTot);
    hipMemsetAsync(denom, 0, (size_t)nhTot * 4, stream);
    hipMemsetAsync(aggr, 0, (size_t)Nn * HC * 4, stream);

    edge_max_k<<<(ehTot + 255) / 256, 256, 0, stream>>>(ei, als, ald, menc, E, ehTot);
    edge_denom_k<<<(ehTot + 255) / 256, 256, 0, stream>>>(ei, als, ald, menc, denom, E, ehTot);

    {
      const long long tot = (long long)nedges * 32;
      edge_aggr_k<<<(int)((tot + 255) / 256), 256, 0, stream>>>(
          ei, als, ald, menc, denom, hbuf, aggr, E, nedges);
    }

    bias_elu_k<<<(Nn * HC + 255) / 256, 256, 0, stream>>>(aggr, bl, act, Nn * HC);
  }

  hipMemsetAsync(g, 0, (size_t)Bb * HC * 4, stream);
  pool_k<<<(Nn * HC + 255) / 256, 256, 0, stream>>>(act, batch, g, Nn * HC);
  mlp_k<<<Bb, 32, 0, stream>>>(g, l1w, l1b, l2w, l2b, out);
}